// Attention_4234837754291
// MI455X (gfx1250) — compile-verified
//
#include <hip/hip_runtime.h>

// ---------------------------------------------------------------------------
// RWKV-v4 block for MI455X (gfx1250, wave32).
//   Stage 1: fused time-shift-mix + 3 GEMMs (k,v,r), f32 WMMA, async-to-LDS
//            double-buffered tiles (ASYNCcnt pipeline).
//   Stage 2: 3-pass chunked log-space WKV scan (16x parallelism vs naive).
//   Stage 3: y = wkv*sigmoid(r) fused into scan pass 3, then output GEMM.
// ---------------------------------------------------------------------------

typedef float v2f __attribute__((ext_vector_type(2)));
typedef float v8f __attribute__((ext_vector_type(8)));

constexpr int   Bsz     = 4;
constexpr int   Tseq    = 4096;
constexpr int   Dm      = 1024;
constexpr int   NC      = 16;            // scan chunks
constexpr int   TC      = Tseq / NC;     // 256 steps per chunk
constexpr float EPSf    = 1e-8f;
constexpr float LOGINIT = -1e38f;

// --- GEMM LDS geometry (units: floats) -------------------------------------
constexpr int ROWF    = 20;                       // 16 data + 4 pad (16B-aligned rows,
                                                  // r*20 mod 64 distinct -> no bank conflicts)
constexpr int A_ROWS  = 128;                      // M tile
constexpr int W_ROWS  = 64;                       // N tile
constexpr int OFF_ACUR = 0;
constexpr int OFF_APRV = A_ROWS * ROWF;           // 2560
constexpr int OFF_W    = 2 * A_ROWS * ROWF;       // 5120
constexpr int SBUF     = (2 * A_ROWS + W_ROWS) * ROWF;   // 6400 floats / buffer
constexpr int OFF_MIX  = 2 * SBUF;                // 12800 (mix vector, staged once)

__device__ __forceinline__ float logaddexpf_(float a, float b) {
    float m = fmaxf(a, b);
    float d = fminf(a, b) - m;           // <= 0
    return m + log1pf(expf(d));
}

// gfx1250 async global->LDS copy (ASYNCcnt). LDS address = low 32 bits of the
// generic pointer (ISA 10.2: LDS aperture keeps wave-relative offset in [31:0]).
__device__ __forceinline__ void async_b128(unsigned lds_off, const float* g) {
    asm volatile("global_load_async_to_lds_b128 %0, %1, off"
                 :: "v"(lds_off), "v"(g) : "memory");
}
__device__ __forceinline__ void wait_async0() {
    asm volatile("s_wait_asynccnt 0x0" ::: "memory");
}

// ---------------------------------------------------------------------------
// GEMM: C[m,n] = sum_j A'[m,j] * W[n,j]
//   MIX=true :  A'[m,j] = mix[j]*A[m,j] + (1-mix[j])*A[m-1,j]   (0 at t==0)
//   MIX=false:  A' = A
// Block: 256 threads (8 wave32), tile 128M x 64N, K-step 16, double-buffered
// async-to-LDS staging. Wave (wm,wn) in 4x2 grid computes 32M x 32N:
// 4 v8f accumulators, 2 A-frags x 2 B-frags -> 4 WMMAs per K-4 step.
// gridDim.z selects the (W, mix, C) triple so k/v/r run in one launch.
// ---------------------------------------------------------------------------
template<bool MIX>
__global__ __launch_bounds__(256)
void gemm_wmma_async(const float* __restrict__ A,
                     const float* __restrict__ W0, const float* __restrict__ W1,
                     const float* __restrict__ W2,
                     const float* __restrict__ mix0, const float* __restrict__ mix1,
                     const float* __restrict__ mix2,
                     float* __restrict__ C0, float* __restrict__ C1,
                     float* __restrict__ C2)
{
    const float* W;  const float* mix;  float* C;
    if      (blockIdx.z == 0) { W = W0; mix = mix0; C = C0; }
    else if (blockIdx.z == 1) { W = W1; mix = mix1; C = C1; }
    else                      { W = W2; mix = mix2; C = C2; }

    __shared__ float sm[2 * SBUF + Dm];            // 2 tile buffers + mix vector

    const int tid   = threadIdx.x;
    const int tileM = blockIdx.y * A_ROWS;
    const int tileN = blockIdx.x * W_ROWS;

    // ---- stage mix vector into LDS once ----
    if constexpr (MIX) {
        const float4 mxv = *(const float4*)(mix + tid * 4);
        *(float4*)&sm[OFF_MIX + tid * 4] = mxv;
    }

    // ---- loader mapping: thread -> (row, 16B chunk) ----
    const int lrow = tid >> 2;                     // 0..63
    const int lcol = tid & 3;                      // chunk 0..3 within 16-float row
    const int mA0  = tileM + lrow;                 // this thread's A rows
    const int mA1  = tileM + lrow + 64;
    const int nW   = tileN + lrow;
    const bool z0  = ((mA0 & (Tseq - 1)) == 0);    // t==0 rows (batch start)
    const bool z1  = ((mA1 & (Tseq - 1)) == 0);
    const int mP0  = z0 ? mA0 : mA0 - 1;           // clamped prev row (no OOB read)
    const int mP1  = z1 ? mA1 : mA1 - 1;

    const float* gA0 = A + (size_t)mA0 * Dm + lcol * 4;
    const float* gA1 = A + (size_t)mA1 * Dm + lcol * 4;
    const float* gP0 = A + (size_t)mP0 * Dm + lcol * 4;
    const float* gP1 = A + (size_t)mP1 * Dm + lcol * 4;
    const float* gW  = W + (size_t)nW  * Dm + lcol * 4;

    const unsigned smBase = (unsigned)(uintptr_t)&sm[0];
    unsigned ldsA0[2], ldsA1[2], ldsP0[2], ldsP1[2], ldsWc[2];
    #pragma unroll
    for (int b = 0; b < 2; ++b) {
        const unsigned base = smBase + (unsigned)(b * SBUF) * 4u;
        ldsA0[b] = base + (unsigned)(OFF_ACUR + lrow        * ROWF) * 4u + lcol * 16u;
        ldsA1[b] = base + (unsigned)(OFF_ACUR + (lrow + 64) * ROWF) * 4u + lcol * 16u;
        ldsP0[b] = base + (unsigned)(OFF_APRV + lrow        * ROWF) * 4u + lcol * 16u;
        ldsP1[b] = base + (unsigned)(OFF_APRV + (lrow + 64) * ROWF) * 4u + lcol * 16u;
        ldsWc[b] = base + (unsigned)(OFF_W    + lrow        * ROWF) * 4u + lcol * 16u;
    }

    auto issue = [&](int b, int kk) {
        async_b128(ldsA0[b], gA0 + kk);
        async_b128(ldsA1[b], gA1 + kk);
        if constexpr (MIX) {
            async_b128(ldsP0[b], gP0 + kk);
            async_b128(ldsP1[b], gP1 + kk);
        }
        async_b128(ldsWc[b], gW + kk);
    };
    // overwrite clamped prev rows with zeros (after wait_async, before barrier)
    auto zerofix = [&](int b) {
        if constexpr (MIX) {
            if (z0) *(float4*)&sm[b * SBUF + OFF_APRV + lrow        * ROWF + lcol * 4] =
                        make_float4(0.f, 0.f, 0.f, 0.f);
            if (z1) *(float4*)&sm[b * SBUF + OFF_APRV + (lrow + 64) * ROWF + lcol * 4] =
                        make_float4(0.f, 0.f, 0.f, 0.f);
        }
    };

    // ---- compute mapping ----
    const int wid  = tid >> 5;
    const int wm   = wid >> 1;            // 0..3 -> M offset wm*32
    const int wn   = wid & 1;             // 0..1 -> N offset wn*32
    const int lane = tid & 31;
    const int half = lane >> 4;           // K pair select within frag
    const int l15  = lane & 15;
    const int ar0  = wm * 32 + l15;       // A tile row (mi=0); +16 for mi=1
    const int bn0  = wn * 32 + l15;       // W tile row (ni=0); +16 for ni=1

    v8f acc00 = {}, acc01 = {}, acc10 = {}, acc11 = {};

    // ---- prologue: fill buffer 0 ----
    issue(0, 0);
    wait_async0();
    zerofix(0);
    __syncthreads();

    int p = 0;
    for (int kk = 0; kk < Dm; kk += 16) {
        if (kk + 16 < Dm) issue(p ^ 1, kk + 16);   // prefetch next slab (overlaps WMMA)

        const float* bufA = &sm[p * SBUF + OFF_ACUR];
        const float* bufP = &sm[p * SBUF + OFF_APRV];
        const float* bufW = &sm[p * SBUF + OFF_W];

        #pragma unroll
        for (int k4 = 0; k4 < 4; ++k4) {
            const int kb = k4 * 4 + half * 2;
            v2f a0, a1;
            if constexpr (MIX) {
                const float mxx = sm[OFF_MIX + kk + kb];
                const float mxy = sm[OFF_MIX + kk + kb + 1];
                float c, pv;
                c = bufA[ar0 * ROWF + kb];            pv = bufP[ar0 * ROWF + kb];
                a0.x = pv + mxx * (c - pv);
                c = bufA[ar0 * ROWF + kb + 1];        pv = bufP[ar0 * ROWF + kb + 1];
                a0.y = pv + mxy * (c - pv);
                c = bufA[(ar0 + 16) * ROWF + kb];     pv = bufP[(ar0 + 16) * ROWF + kb];
                a1.x = pv + mxx * (c - pv);
                c = bufA[(ar0 + 16) * ROWF + kb + 1]; pv = bufP[(ar0 + 16) * ROWF + kb + 1];
                a1.y = pv + mxy * (c - pv);
            } else {
                a0.x = bufA[ar0 * ROWF + kb];
                a0.y = bufA[ar0 * ROWF + kb + 1];
                a1.x = bufA[(ar0 + 16) * ROWF + kb];
                a1.y = bufA[(ar0 + 16) * ROWF + kb + 1];
            }
            v2f b0, b1;
            b0.x = bufW[bn0 * ROWF + kb];
            b0.y = bufW[bn0 * ROWF + kb + 1];
            b1.x = bufW[(bn0 + 16) * ROWF + kb];
            b1.y = bufW[(bn0 + 16) * ROWF + kb + 1];

            acc00 = __builtin_amdgcn_wmma_f32_16x16x4_f32(false, a0, false, b0, (short)0, acc00, false, false);
            acc01 = __builtin_amdgcn_wmma_f32_16x16x4_f32(false, a0, false, b1, (short)0, acc01, false, false);
            acc10 = __builtin_amdgcn_wmma_f32_16x16x4_f32(false, a1, false, b0, (short)0, acc10, false, false);
            acc11 = __builtin_amdgcn_wmma_f32_16x16x4_f32(false, a1, false, b1, (short)0, acc11, false, false);
        }

        wait_async0();          // next slab landed in LDS
        zerofix(p ^ 1);
        __syncthreads();        // everyone done reading buf p; next data visible
        p ^= 1;
    }

    // ---- store C: VGPR e -> M = e + 8*half, N = l15 ----
    const int crow = tileM + wm * 32 + half * 8;
    const int ccol = tileN + wn * 32 + l15;
    #pragma unroll
    for (int e = 0; e < 8; ++e) {
        C[(size_t)(crow + e)      * Dm + ccol]      = acc00[e];
        C[(size_t)(crow + e)      * Dm + ccol + 16] = acc01[e];
        C[(size_t)(crow + e + 16) * Dm + ccol]      = acc10[e];
        C[(size_t)(crow + e + 16) * Dm + ccol + 16] = acc11[e];
    }
}

// ---------------------------------------------------------------------------
// WKV scan, pass 1: per-chunk states from -inf init. One thread per (b,c,d).
// ---------------------------------------------------------------------------
__global__ __launch_bounds__(256)
void wkv_pass1(const float* __restrict__ kbuf, const float* __restrict__ vbuf,
               const float* __restrict__ time_decay,
               float* __restrict__ Ap, float* __restrict__ Am, float* __restrict__ Ab)
{
    const int gid = blockIdx.x * 256 + threadIdx.x;      // 0 .. B*NC*D-1
    const int d = gid & (Dm - 1);
    const int c = (gid >> 10) & (NC - 1);
    const int b = gid >> 14;
    const float w = expf(time_decay[d]);

    float ap = LOGINIT, am = LOGINIT, bb = LOGINIT;
    size_t idx = ((size_t)b * Tseq + (size_t)c * TC) * Dm + d;
    for (int t = 0; t < TC; ++t) {
        const float kt = kbuf[idx];
        const float vt = vbuf[idx];
        const float lvp = logf(fmaxf(vt, 0.f) + EPSf);
        const float lvm = logf(fmaxf(-vt, 0.f) + EPSf);
        ap = logaddexpf_(ap - w, kt + lvp);
        am = logaddexpf_(am - w, kt + lvm);
        bb = logaddexpf_(bb - w, kt);
        idx += Dm;
    }
    const int sidx = (b * NC + c) * Dm + d;
    Ap[sidx] = ap; Am[sidx] = am; Ab[sidx] = bb;
}

// ---------------------------------------------------------------------------
// WKV scan, pass 2: combine chunk states (exact composition: state after n
// steps with init s0 is logaddexp(s0 - n*w, A_chunk)). One thread per (b,d).
// ---------------------------------------------------------------------------
__global__ __launch_bounds__(256)
void wkv_pass2(const float* __restrict__ Ap, const float* __restrict__ Am,
               const float* __restrict__ Ab, const float* __restrict__ time_decay,
               float* __restrict__ Ip, float* __restrict__ Im, float* __restrict__ Ib)
{
    const int gid = blockIdx.x * 256 + threadIdx.x;      // 0 .. B*D-1
    const int d = gid & (Dm - 1);
    const int b = gid >> 10;
    const float w  = expf(time_decay[d]);
    const float wc = w * (float)TC;

    float sp = LOGINIT, sm = LOGINIT, sb = LOGINIT;
    for (int c = 0; c < NC; ++c) {
        const int sidx = (b * NC + c) * Dm + d;
        Ip[sidx] = sp; Im[sidx] = sm; Ib[sidx] = sb;     // init *before* chunk c
        sp = logaddexpf_(sp - wc, Ap[sidx]);
        sm = logaddexpf_(sm - wc, Am[sidx]);
        sb = logaddexpf_(sb - wc, Ab[sidx]);
    }
}

// ---------------------------------------------------------------------------
// WKV scan, pass 3: per-step outputs per chunk with correct init state.
// Fuses y = wkv * sigmoid(r); writes y in place over the r buffer.
// ---------------------------------------------------------------------------
__global__ __launch_bounds__(256)
void wkv_pass3(const float* __restrict__ kbuf, const float* __restrict__ vbuf,
               float* __restrict__ rbuf,
               const float* __restrict__ time_decay, const float* __restrict__ time_first,
               const float* __restrict__ Ip, const float* __restrict__ Im,
               const float* __restrict__ Ib)
{
    const int gid = blockIdx.x * 256 + threadIdx.x;      // 0 .. B*NC*D-1
    const int d = gid & (Dm - 1);
    const int c = (gid >> 10) & (NC - 1);
    const int b = gid >> 14;
    const float w = expf(time_decay[d]);
    const float u = time_first[d];

    const int sidx = (b * NC + c) * Dm + d;
    float ap = Ip[sidx], am = Im[sidx], bb = Ib[sidx];

    size_t idx = ((size_t)b * Tseq + (size_t)c * TC) * Dm + d;
    for (int t = 0; t < TC; ++t) {
        const float kt = kbuf[idx];
        const float vt = vbuf[idx];
        const float rt = rbuf[idx];
        const float lvp = logf(fmaxf(vt, 0.f) + EPSf);
        const float lvm = logf(fmaxf(-vt, 0.f) + EPSf);
        const float uk  = u + kt;
        const float lden = logaddexpf_(uk, bb);
        const float wkv = expf(logaddexpf_(uk + lvp, ap) - lden)
                        - expf(logaddexpf_(uk + lvm, am) - lden);
        rbuf[idx] = wkv * (1.f / (1.f + expf(-rt)));     // y = wkv * sigmoid(r)
        ap = logaddexpf_(ap - w, kt + lvp);
        am = logaddexpf_(am - w, kt + lvm);
        bb = logaddexpf_(bb - w, kt);
        idx += Dm;
    }
}

// ---------------------------------------------------------------------------
extern "C" void kernel_launch(void* const* d_in, const int* in_sizes, int n_in,
                              void* d_out, int out_size, void* d_ws, size_t ws_size,
                              hipStream_t stream)
{
    (void)in_sizes; (void)n_in; (void)out_size; (void)ws_size;

    const float* x  = (const float*)d_in[0];
    const float* td = (const float*)d_in[1];   // time_decay [D]
    const float* tf = (const float*)d_in[2];   // time_first [D]
    const float* mk = (const float*)d_in[3];   // time_mix_k [D]
    const float* mv = (const float*)d_in[4];
    const float* mr = (const float*)d_in[5];
    const float* Wk = (const float*)d_in[6];   // [D,D]
    const float* Wv = (const float*)d_in[7];
    const float* Wr = (const float*)d_in[8];
    const float* Wo = (const float*)d_in[9];
    float* out = (float*)d_out;

    const size_t BTD = (size_t)Bsz * Tseq * Dm;      // 16.8M floats per buffer
    const size_t SST = (size_t)Bsz * NC * Dm;        // chunk-state stride
    float* ws = (float*)d_ws;
    float* kb = ws;
    float* vb = ws + BTD;
    float* rb = ws + 2 * BTD;                        // r, then y in-place
    float* Ap = ws + 3 * BTD;
    float* Am = Ap + SST;
    float* Ab = Am + SST;
    float* Ip = Ab + SST;
    float* Im = Ip + SST;
    float* Ib = Im + SST;

    // Stage 1: k, v, r GEMMs with fused time-shift mix (z = projection)
    dim3 gKVR(Dm / W_ROWS, (Bsz * Tseq) / A_ROWS, 3);
    gemm_wmma_async<true><<<gKVR, 256, 0, stream>>>(
        x, Wk, Wv, Wr, mk, mv, mr, kb, vb, rb);

    // Stage 2: chunked log-space WKV scan
    wkv_pass1<<<(Bsz * NC * Dm) / 256, 256, 0, stream>>>(kb, vb, td, Ap, Am, Ab);
    wkv_pass2<<<(Bsz * Dm) / 256, 256, 0, stream>>>(Ap, Am, Ab, td, Ip, Im, Ib);
    wkv_pass3<<<(Bsz * NC * Dm) / 256, 256, 0, stream>>>(kb, vb, rb, td, tf, Ip, Im, Ib);

    // Stage 3: out = y @ W_output^T
    dim3 gOut(Dm / W_ROWS, (Bsz * Tseq) / A_ROWS, 1);
    gemm_wmma_async<false><<<gOut, 256, 0, stream>>>(
        rb, Wo, Wo, Wo, nullptr, nullptr, nullptr, out, out, out);
}